// SAGPool_69320772157743
// MI455X (gfx1250) — compile-verified
//
#include <hip/hip_runtime.h>
#include <hip/hip_bf16.h>
#include <math.h>

// Problem constants (from reference)
#define G        256
#define NPG      1024
#define N_TOTAL  (G * NPG)            // 262144
#define AVG_DEG  32
#define E_TOTAL  (N_TOTAL * AVG_DEG)  // 8388608
#define D_FEAT   128
#define K_KEEP   820                  // ceil(0.8 * 1024)
#define NK       (G * K_KEEP)         // 209920

typedef float v2f __attribute__((ext_vector_type(2)));
typedef float v8f __attribute__((ext_vector_type(8)));

// ---------------------------------------------------------------- init
__global__ void init_kernel(float* __restrict__ deg, float* __restrict__ agg,
                            int* __restrict__ node_map) {
    int i = blockIdx.x * blockDim.x + threadIdx.x;
    if (i < N_TOTAL) {
        deg[i] = 0.0f;
        agg[i] = 0.0f;
        node_map[i] = -1;
    }
}

// ---------------------------------------------------------------- GEMV via f32 WMMA
// xw[N] = x[N,128] @ W[128]. One wave handles 16 rows.
// Stage the 16x128 f32 tile into LDS with GLOBAL_LOAD_ASYNC_TO_LDS_B128
// (coalesced 512B per row, ASYNCcnt-tracked), then feed WMMA A-fragments
// from LDS. Row stride padded to 132 floats: lane r<16 reads banks
// {4r+k,4r+k+1}, lane 16+r reads {4r+k+2,4r+k+3} -> all 64 banks, no conflict.
// A tile (16x4 f32): lane l<16 -> row l, K+0/K+1 ; lane l>=16 -> row l-16, K+2/K+3.
// B (4x16) broadcasts W[k] across all 16 columns -> every column of D equals xw.
#define GEMV_WAVES 4
#define ROW_PITCH  132

__global__ __launch_bounds__(32 * GEMV_WAVES) void gemv_wmma_kernel(
        const float* __restrict__ x, const float* __restrict__ W,
        float* __restrict__ xw) {
    __shared__ float tile[GEMV_WAVES][16 * ROW_PITCH];

    int wlocal = threadIdx.x >> 5;
    int lane   = threadIdx.x & 31;
    int wave   = blockIdx.x * GEMV_WAVES + wlocal;
    int row_base = wave * 16;
    if (row_base >= N_TOTAL) return;   // uniform per wave: EXEC stays all-1s

    // ---- async stage: 16 rows x 512B, one b128 per lane per row ----
    unsigned lds_base = (unsigned)(size_t)(&tile[wlocal][0]);   // low 32b of flat = LDS offset
    #pragma unroll
    for (int p = 0; p < 16; ++p) {
        unsigned loff = lds_base + (unsigned)(p * ROW_PITCH + lane * 4) * 4u;
        int      goff = (row_base + p) * (D_FEAT * 4) + lane * 16;   // byte offset
        asm volatile("global_load_async_to_lds_b128 %0, %1, %2"
                     :: "v"(loff), "v"(goff), "s"(x) : "memory");
    }
    asm volatile("s_wait_asynccnt 0x0" ::: "memory");

    int m    = lane & 15;
    int koff = (lane >> 4) << 1;              // 0 for lanes 0-15, 2 for lanes 16-31
    const float* arow = &tile[wlocal][m * ROW_PITCH];

    v8f c = {};
    #pragma unroll 4
    for (int k = 0; k < D_FEAT; k += 4) {
        v2f a, b;
        a.x = arow[k + koff];                 // ds_load_b64, conflict-free
        a.y = arow[k + koff + 1];
        b.x = W[k + koff];                    // 512B, WGP$-resident
        b.y = W[k + koff + 1];
        c = __builtin_amdgcn_wmma_f32_16x16x4_f32(
                /*neg_a=*/false, a, /*neg_b=*/false, b,
                /*c_mod=*/(short)0, c, /*reuse_a=*/false, /*reuse_b=*/false);
    }
    // C/D layout: VGPR r -> M=r (lanes 0-15), M=r+8 (lanes 16-31); all columns equal.
    if (lane == 0) {
        #pragma unroll
        for (int r = 0; r < 8; ++r) xw[row_base + r] = c[r];
    } else if (lane == 16) {
        #pragma unroll
        for (int r = 0; r < 8; ++r) xw[row_base + 8 + r] = c[r];
    }
}

// ---------------------------------------------------------------- degree
__global__ void deg_kernel(const int* __restrict__ dst, float* __restrict__ deg) {
    int e = blockIdx.x * blockDim.x + threadIdx.x;
    if (e < E_TOTAL) atomicAdd(&deg[dst[e]], 1.0f);
}

__global__ void dinv_kernel(const float* __restrict__ deg, float* __restrict__ dinv) {
    int i = blockIdx.x * blockDim.x + threadIdx.x;
    if (i < N_TOTAL) dinv[i] = rsqrtf(deg[i] + 1.0f);
}

// ---------------------------------------------------------------- normalized scatter-add
__global__ void agg_kernel(const int* __restrict__ src, const int* __restrict__ dst,
                           const float* __restrict__ dinv, const float* __restrict__ xw,
                           float* __restrict__ agg) {
    int e = blockIdx.x * blockDim.x + threadIdx.x;
    if (e < E_TOTAL) {
        int s = src[e];
        int d = dst[e];
        atomicAdd(&agg[d], dinv[s] * dinv[d] * xw[s]);
    }
}

__global__ void score_kernel(const float* __restrict__ agg, const float* __restrict__ xw,
                             const float* __restrict__ dinv, const float* __restrict__ b,
                             float* __restrict__ score) {
    int i = blockIdx.x * blockDim.x + threadIdx.x;
    if (i < N_TOTAL) {
        float di = dinv[i];
        score[i] = agg[i] + xw[i] * di * di + b[0];
    }
}

// ---------------------------------------------------------------- per-graph top-k (bitonic)
// key = ordered_uint(score) << 32 | (1023 - idx): descending sort gives descending
// score with ascending-index tie-break (matches jax.lax.top_k).
__global__ __launch_bounds__(1024) void topk_kernel(
        const float* __restrict__ score,
        int* __restrict__ perm_i, float* __restrict__ tanhsc,
        int* __restrict__ node_map,
        float* __restrict__ out_perm, float* __restrict__ out_batch) {
    __shared__ unsigned long long keys[NPG];
    int g = blockIdx.x;
    int t = threadIdx.x;

    float s = score[g * NPG + t];
    unsigned u   = __float_as_uint(s);
    unsigned ord = (u & 0x80000000u) ? ~u : (u | 0x80000000u);
    keys[t] = ((unsigned long long)ord << 32) | (unsigned)(NPG - 1 - t);
    __syncthreads();

    for (int k = 2; k <= NPG; k <<= 1) {
        for (int j = k >> 1; j > 0; j >>= 1) {
            int ixj = t ^ j;
            if (ixj > t) {
                unsigned long long a = keys[t];
                unsigned long long c = keys[ixj];
                bool up  = ((t & k) == 0);
                bool swp = up ? (a < c) : (a > c);   // descending overall
                if (swp) { keys[t] = c; keys[ixj] = a; }
            }
            __syncthreads();
        }
    }

    if (t < K_KEEP) {
        unsigned long long kk = keys[t];
        int local = NPG - 1 - (int)(kk & 0xFFFFFFFFull);
        int gnode = g * NPG + local;
        int pos   = g * K_KEEP + t;
        perm_i[pos]     = gnode;
        tanhsc[pos]     = tanhf(score[gnode]);
        node_map[gnode] = pos;
        out_perm[pos]   = (float)gnode;
        out_batch[pos]  = (float)g;                 // batch[perm] == graph id
    }
}

// ---------------------------------------------------------------- row gather + tanh scale
// One wave per kept row; 32 lanes x float4 = 128 floats (b128 loads/stores).
__global__ void gather_kernel(const float* __restrict__ x,
                              const int* __restrict__ perm_i,
                              const float* __restrict__ tanhsc,
                              float* __restrict__ x_out, float* __restrict__ x_ae) {
    int wave = (blockIdx.x * blockDim.x + threadIdx.x) >> 5;
    int lane = threadIdx.x & 31;
    if (wave >= NK) return;
    int row  = perm_i[wave];
    float tv = tanhsc[wave];
    const float4* srcp = (const float4*)(x + (size_t)row * D_FEAT);
    float4 v = srcp[lane];
    ((float4*)(x_ae  + (size_t)wave * D_FEAT))[lane] = v;
    float4 w; w.x = v.x * tv; w.y = v.y * tv; w.z = v.z * tv; w.w = v.w * tv;
    ((float4*)(x_out + (size_t)wave * D_FEAT))[lane] = w;
}

// ---------------------------------------------------------------- edge filter / remap
__global__ void edge_kernel(const int* __restrict__ ei, const int* __restrict__ node_map,
                            float* __restrict__ out_src, float* __restrict__ out_dst,
                            float* __restrict__ out_mask) {
    int e = blockIdx.x * blockDim.x + threadIdx.x;
    if (e >= E_TOTAL) return;
    int s = ei[e];
    int d = ei[E_TOTAL + e];
    int ms = node_map[s];
    int md = node_map[d];
    bool keep = (ms >= 0) && (md >= 0);
    out_src[e]  = (float)(keep ? ms : -1);
    out_dst[e]  = (float)(keep ? md : -1);
    out_mask[e] = keep ? 1.0f : 0.0f;
}

// ---------------------------------------------------------------- launch
extern "C" void kernel_launch(void* const* d_in, const int* in_sizes, int n_in,
                              void* d_out, int out_size, void* d_ws, size_t ws_size,
                              hipStream_t stream) {
    const float* x  = (const float*)d_in[0];       // [N, 128]
    const int*   ei = (const int*)  d_in[1];       // [2, E]: src row then dst row
    const float* W  = (const float*)d_in[3];       // [128]
    const float* b  = (const float*)d_in[4];       // [1]

    // workspace partition
    char* p = (char*)d_ws;
    float* xw       = (float*)p; p += (size_t)N_TOTAL * sizeof(float);
    float* deg      = (float*)p; p += (size_t)N_TOTAL * sizeof(float);
    float* dinv     = (float*)p; p += (size_t)N_TOTAL * sizeof(float);
    float* agg      = (float*)p; p += (size_t)N_TOTAL * sizeof(float);
    float* score    = (float*)p; p += (size_t)N_TOTAL * sizeof(float);
    int*   node_map = (int*)  p; p += (size_t)N_TOTAL * sizeof(int);
    int*   perm_i   = (int*)  p; p += (size_t)NK * sizeof(int);
    float* tanhsc   = (float*)p; p += (size_t)NK * sizeof(float);

    // output partition (reference tuple order, all as f32)
    float* out      = (float*)d_out;
    float* out_x    = out;                                   // [NK, 128]
    float* out_esrc = out_x    + (size_t)NK * D_FEAT;        // [E]
    float* out_edst = out_esrc + (size_t)E_TOTAL;            // [E]
    float* out_mask = out_edst + (size_t)E_TOTAL;            // [E]
    float* out_bat  = out_mask + (size_t)E_TOTAL;            // [NK]
    float* out_perm = out_bat  + (size_t)NK;                 // [NK]
    float* out_ae   = out_perm + (size_t)NK;                 // [NK, 128]

    const int* esrc = ei;
    const int* edst = ei + E_TOTAL;

    init_kernel<<<N_TOTAL / 256, 256, 0, stream>>>(deg, agg, node_map);

    // N/16 waves; GEMV_WAVES waves per block
    gemv_wmma_kernel<<<(N_TOTAL / 16) / GEMV_WAVES, 32 * GEMV_WAVES, 0, stream>>>(x, W, xw);

    deg_kernel<<<E_TOTAL / 256, 256, 0, stream>>>(edst, deg);
    dinv_kernel<<<N_TOTAL / 256, 256, 0, stream>>>(deg, dinv);
    agg_kernel<<<E_TOTAL / 256, 256, 0, stream>>>(esrc, edst, dinv, xw, agg);
    score_kernel<<<N_TOTAL / 256, 256, 0, stream>>>(agg, xw, dinv, b, score);

    topk_kernel<<<G, 1024, 0, stream>>>(score, perm_i, tanhsc, node_map,
                                        out_perm, out_bat);

    gather_kernel<<<(NK * 32) / 256, 256, 0, stream>>>(x, perm_i, tanhsc,
                                                       out_x, out_ae);

    edge_kernel<<<E_TOTAL / 256, 256, 0, stream>>>(ei, node_map,
                                                   out_esrc, out_edst, out_mask);
}